// IterativeGaussianKDE_10831907520619
// MI455X (gfx1250) — compile-verified
//
#include <hip/hip_runtime.h>
#include <math.h>

typedef __attribute__((ext_vector_type(2))) float v2f;
typedef __attribute__((ext_vector_type(8))) float v8f;

// ---------------------------------------------------------------------------
// Pass 1: reduce each image's mask to a {0,1} float flag in workspace.
// (masks are all-ones in the harness, but the reference zeroes empty-mask
//  images, so reproduce that faithfully; d_ws is poisoned -> write every call)
// ---------------------------------------------------------------------------
__global__ void kde_mask_flags(const float* __restrict__ masks,
                               float* __restrict__ flags, int n)
{
    __shared__ int s_any;
    if (threadIdx.x == 0) s_any = 0;
    __syncthreads();
    const float* m = masks + (size_t)blockIdx.x * (size_t)n;
    int any = 0;
    for (int i = threadIdx.x; i < n; i += blockDim.x)
        any |= (m[i] != 0.0f) ? 1 : 0;
    if (any) atomicOr(&s_any, 1);
    __syncthreads();
    if (threadIdx.x == 0) flags[blockIdx.x] = s_any ? 1.0f : 0.0f;
}

// ---------------------------------------------------------------------------
// Pass 2: KDE via V_WMMA_F32_16X16X4_F32.
//   D[m][n] = sum_k A[m][k]*B[k][n] + C[m][n]
// with A[m] = [x0,x1,x2,||x||^2], B[:,n] = [2s*c0, 2s*c1, 2s*c2, -s],
//      C[m][n] = -s*||c_n||^2 + log2(CONST1/npix),  s = log2(e)/CONST2
// so  D = -s*||x_m - c_n||^2 + log2(CONST1/npix)  and the per-pair tail is
// exactly one v_exp_f32 + one v_add_f32.
//
// One wave <-> one (image, 16-center tile); loops over npix/16 pixel tiles.
// ---------------------------------------------------------------------------
__global__ void __launch_bounds__(256)
kde_wmma_kernel(const float* __restrict__ images,
                const float* __restrict__ cspace,
                const float* __restrict__ flags,
                float* __restrict__ out,
                int npix, int ktiles_per_img,
                float two_s, float neg_s, float bias_log)
{
    const int lane  = (int)(threadIdx.x & 31u);
    const int wave  = (int)(threadIdx.x >> 5u);
    const int wid   = (int)blockIdx.x * ((int)blockDim.x >> 5) + wave;
    const int img   = wid / ktiles_per_img;
    const int ktile = wid % ktiles_per_img;
    const int half  = lane >> 4;   // 0: lanes 0-15, 1: lanes 16-31
    const int ncol  = lane & 15;   // column n of the C/D tile held by this lane

    // ---- loop-invariant B fragment + C bias from the kernel centers ----
    const int   kidx = ktile * 16 + ncol;
    const float c0 = cspace[kidx * 3 + 0];
    const float c1 = cspace[kidx * 3 + 1];
    const float c2 = cspace[kidx * 3 + 2];
    const float cn = c0 * c0 + c1 * c1 + c2 * c2;

    // B 4x16 f32 layout: vgpr0 = row K=0 (lanes 0-15) / K=2 (lanes 16-31),
    //                    vgpr1 = row K=1 (lanes 0-15) / K=3 (lanes 16-31)
    v2f bfrag;
    bfrag.x = half ? (two_s * c2) : (two_s * c0);
    bfrag.y = half ? neg_s        : (two_s * c1);

    const float cbias = neg_s * cn + bias_log;
    v8f cfrag;
#pragma unroll
    for (int j = 0; j < 8; ++j) cfrag[j] = cbias;   // column-constant bias

    const float* __restrict__ xbase =
        images + (size_t)img * (size_t)npix * 3u;

    v8f acc = {};
    const int ntiles = npix >> 4;
#pragma unroll 2
    for (int t = 0; t < ntiles; ++t) {
        // A 16x4 f32 layout: lanes 0-15 hold row M=lane, (K0,K1) in (v0,v1);
        //                    lanes 16-31 hold row M=lane-16, (K2,K3).
        const float* xr = xbase + (size_t)(t * 16 + ncol) * 3u;
        const float f0 = xr[0], f1 = xr[1], f2 = xr[2];
        const float xn = f0 * f0 + f1 * f1 + f2 * f2;
        v2f afrag;
        afrag.x = half ? f2 : f0;
        afrag.y = half ? xn : f1;

        v8f d = __builtin_amdgcn_wmma_f32_16x16x4_f32(
            /*neg_a=*/false, afrag, /*neg_b=*/false, bfrag,
            /*c_mod=*/(short)0, cfrag, /*reuse_a=*/false, /*reuse_b=*/false);

#pragma unroll
        for (int j = 0; j < 8; ++j)
            acc[j] += __builtin_amdgcn_exp2f(d[j]);
    }

    // rows m = j + 8*half accumulated per lane; fold 8 vgprs, then the two
    // lane halves (SWAPX16 ds_swizzle: xor=0x10, and=0x1f -> imm 0x401F).
    float s = (((acc[0] + acc[1]) + (acc[2] + acc[3])) +
               ((acc[4] + acc[5]) + (acc[6] + acc[7])));
    int   sw  = __builtin_amdgcn_ds_swizzle(__float_as_int(s), 0x401F);
    float tot = s + __int_as_float(sw);

    if (half == 0) {
        out[(size_t)img * (size_t)(ktiles_per_img * 16) + (size_t)kidx] =
            tot * flags[img];
    }
}

// ---------------------------------------------------------------------------
extern "C" void kernel_launch(void* const* d_in, const int* in_sizes, int n_in,
                              void* d_out, int out_size, void* d_ws, size_t ws_size,
                              hipStream_t stream)
{
    const float* images = (const float*)d_in[0]; // [2,3,128,128] f32
    const float* masks  = (const float*)d_in[1]; // [2,1,128,128] f32
    const float* cspace = (const float*)d_in[2]; // [32768,3]     f32

    const int b       = 2;
    const int npix    = in_sizes[0] / (3 * b);   // 16384 pixels per image
    const int nmask   = in_sizes[1] / b;         // 16384 mask elems per image
    const int K       = in_sizes[2] / 3;         // 32768 kernel centers
    const int ktiles  = K / 16;                  // 2048 column tiles

    // constants (host-side plain math: graph-capture safe)
    const double C1   = sqrt(2000.0) * pow(2.0 * M_PI, -1.5);
    const double s    = M_LOG2E / 4000.0;        // log2(e)/CONST2
    const float  two_s    = (float)(2.0 * s);
    const float  neg_s    = (float)(-s);
    const float  bias_log = (float)log2(C1 / (double)npix);

    float* flags = (float*)d_ws;                 // 2 floats of scratch
    float* out   = (float*)d_out;                // [2, 32768] f32

    kde_mask_flags<<<b, 256, 0, stream>>>(masks, flags, nmask);

    const int waves  = b * ktiles;               // 4096 wave32s
    const int blocks = waves / 8;                // 256 threads = 8 waves/block
    kde_wmma_kernel<<<blocks, 256, 0, stream>>>(
        images, cspace, flags, out, npix, ktiles, two_s, neg_s, bias_log);
}